// GraphEncoder_55301998903770
// MI455X (gfx1250) — compile-verified
//
#include <hip/hip_runtime.h>
#include <math.h>

#define B_     128
#define N_     2048
#define F_IN_  17
#define KPAD_  20
#define D_NODE_ 128
#define D_GAT_ 96
#define D_OUT_ 256
#define LN_EPS_ 1e-5f
#define NEG_SLOPE_ 0.2f

typedef __attribute__((ext_vector_type(2))) float v2f;
typedef __attribute__((ext_vector_type(4))) float v4f;
typedef __attribute__((ext_vector_type(8))) float v8f;

__device__ __forceinline__ float gelu_f(float x) {
    return 0.5f * x * (1.0f + erff(x * 0.70710678118654752f));
}
__device__ __forceinline__ float lrelu_f(float x) {
    return x > 0.0f ? x : NEG_SLOPE_ * x;
}

// LayerNorm on a 16 x W fp32 tile in LDS; one wave. Lane l owns row (l&15),
// half (l>>4) of the channels. Cross-half combine via lane shuffle xor 16.
__device__ __forceinline__ void ln_tile(float* t, int W, const float* g,
                                        const float* bvec, int lane) {
    int row = lane & 15, half = lane >> 4;
    float* rp = t + row * W;
    int c0 = half * (W / 2);
    float s = 0.0f, ss = 0.0f;
    for (int i = 0; i < W / 2; ++i) { float v = rp[c0 + i]; s += v; ss += v * v; }
    s  += __shfl_xor(s, 16, 32);
    ss += __shfl_xor(ss, 16, 32);
    float mu  = s / (float)W;
    float var = ss / (float)W - mu * mu;
    float rs  = rsqrtf(var + LN_EPS_);
    for (int i = 0; i < W / 2; ++i) {
        int c = c0 + i;
        rp[c] = (rp[c] - mu) * rs * g[c] + bvec[c];
    }
}

// ---------------------------------------------------------------------------
// Weight prep: Wt[c*Kpad + k] = (k < K) ? W[k*Ncols + c] : 0
// Column-major + zero-padded K so WMMA B-fragments are one aligned b64 load
// with no bounds guards (kills the exec-mask branch + s_wait_xcnt pattern).
// ---------------------------------------------------------------------------
__global__ __launch_bounds__(256) void k_prep_w(
    const float* __restrict__ W, float* __restrict__ Wt,
    int K, int Kpad, int Ncols) {
    int i = blockIdx.x * 256 + threadIdx.x;
    if (i < Ncols * Kpad) {
        int c = i / Kpad, k = i % Kpad;
        Wt[i] = (k < K) ? W[k * Ncols + c] : 0.0f;
    }
}

// ---------------------------------------------------------------------------
// K1: fused fc_node (17->128) + GELU + LN + conv1 lin_l (128->96).
// One wave per 16-node tile. h1 lives only in LDS; node-0 rows of h1 are
// exported (per batch) for the tiny xr1 GEMV.
// Wnt: [128][20] col-major padded; Wl1t: [96][128] col-major.
// ---------------------------------------------------------------------------
__global__ __launch_bounds__(32) void k_node_xl1(
    const float* __restrict__ x, const float* __restrict__ Wnt,
    const float* __restrict__ bn, const float* __restrict__ lg,
    const float* __restrict__ lb, const float* __restrict__ Wl1t,
    const float* __restrict__ bl1, float* __restrict__ xl1_out,
    float* __restrict__ h0_out) {
    __shared__ float xs[16 * KPAD_];   // x tile, K padded 17 -> 20 with zeros
    __shared__ float h[16 * D_NODE_];  // h1 tile
    __shared__ float o[16 * D_GAT_];   // xl1 out tile (for coalesced stores)
    const int lane = threadIdx.x;
    const int tile = blockIdx.x;
    const int b  = tile / (N_ / 16);
    const int nt = tile % (N_ / 16);
    const int m = lane & 15, hi = lane >> 4;

    const float* xrow = x + (size_t)b * N_ * F_IN_ + (size_t)nt * 16 * F_IN_;
    for (int i = lane; i < 16 * KPAD_; i += 32) xs[i] = 0.0f;
    __syncthreads();
    for (int i = lane; i < 16 * F_IN_; i += 32) {
        int r = i / F_IN_, c = i % F_IN_;
        xs[r * KPAD_ + c] = xrow[i];
    }
    __syncthreads();

    // Stage 1: h = x @ W_node  (M=16, N=128, K=17 padded to 20)
    for (int ntile = 0; ntile < 8; ++ntile) {
        int n0 = ntile * 16;
        const float* wc = Wnt + (size_t)(n0 + m) * KPAD_;
        v8f acc = {};
#pragma unroll
        for (int k0 = 0; k0 < KPAD_; k0 += 4) {
            int ka = k0 + 2 * hi;                      // even -> 8B aligned
            v2f a  = *(const v2f*)&xs[m * KPAD_ + ka];
            v2f bb = *(const v2f*)&wc[ka];
            acc = __builtin_amdgcn_wmma_f32_16x16x4_f32(
                false, a, false, bb, (short)0, acc, false, false);
        }
#pragma unroll
        for (int r = 0; r < 8; ++r) {
            int row = r + 8 * hi, col = n0 + m;
            h[row * D_NODE_ + col] = gelu_f(acc[r] + bn[col]);
        }
    }
    __syncthreads();
    ln_tile(h, D_NODE_, lg, lb, lane);
    __syncthreads();

    if (nt == 0) {  // export h1[b,0,:] for xr1
        for (int i = lane; i < D_NODE_; i += 32) h0_out[b * D_NODE_ + i] = h[i];
    }

    // Stage 2: xl1 = h @ Wl1 + bl1  (M=16, N=96, K=128)
    for (int ntile = 0; ntile < 6; ++ntile) {
        int n0 = ntile * 16;
        const float* wc = Wl1t + (size_t)(n0 + m) * D_NODE_;
        v8f acc = {};
#pragma unroll 8
        for (int k0 = 0; k0 < D_NODE_; k0 += 4) {
            int ka = k0 + 2 * hi;
            v2f a  = *(const v2f*)&h[m * D_NODE_ + ka];
            v2f bb = *(const v2f*)&wc[ka];
            acc = __builtin_amdgcn_wmma_f32_16x16x4_f32(
                false, a, false, bb, (short)0, acc, false, false);
        }
#pragma unroll
        for (int r = 0; r < 8; ++r) {
            int row = r + 8 * hi, col = n0 + m;
            o[row * D_GAT_ + col] = acc[r] + bl1[col];
        }
    }
    __syncthreads();
    float* outp = xl1_out + ((size_t)b * N_ + (size_t)nt * 16) * D_GAT_;
    for (int i = lane * 4; i < 16 * D_GAT_; i += 128)
        *(v4f*)&outp[i] = *(const v4f*)&o[i];
}

// ---------------------------------------------------------------------------
// K2/K5: xr = hrow @ Wr + br for node 0 of each batch (tiny GEMV per batch).
// ---------------------------------------------------------------------------
__global__ __launch_bounds__(128) void k_xr(
    const float* __restrict__ hsrc, int K, const float* __restrict__ W,
    const float* __restrict__ bvec, float* __restrict__ out) {
    const int b = blockIdx.x, tid = threadIdx.x;
    __shared__ float hrow[128];
    if (tid < K) hrow[tid] = hsrc[b * K + tid];
    __syncthreads();
    if (tid < D_GAT_) {
        float acc = bvec[tid];
        for (int k = 0; k < K; ++k) acc += hrow[k] * W[k * D_GAT_ + tid];
        out[b * D_GAT_ + tid] = acc;
    }
}

// ---------------------------------------------------------------------------
// K3/K6: GATv2 attention into node 0 (one block per batch):
//   e_j[h] = sum_c att[h,c] * lrelu(xl[j,h,c] + xr[h,c]); softmax over j;
//   out0 = sum_j alpha_j xl[j]; g0 = LN(gelu(out0 + bias)).
// Scores cached in LDS (2048 x 3 = 24 KB).
// ---------------------------------------------------------------------------
__global__ __launch_bounds__(128) void k_att(
    const float* __restrict__ xl, const float* __restrict__ xr,
    const float* __restrict__ att, const float* __restrict__ bias,
    const float* __restrict__ lg, const float* __restrict__ lb,
    float* __restrict__ g0_out, size_t ostride) {
    const int b = blockIdx.x, tid = threadIdx.x;
    __shared__ float sc[N_ * 3];
    __shared__ float xrs[D_GAT_], atts[D_GAT_];
    __shared__ float red[128];
    __shared__ float mh[3], sh[3];
    if (tid < D_GAT_) { xrs[tid] = xr[b * D_GAT_ + tid]; atts[tid] = att[tid]; }
    __syncthreads();

    const float* X = xl + (size_t)b * N_ * D_GAT_;
    for (int j = tid; j < N_; j += 128) {
        const float* xj = X + (size_t)j * D_GAT_;
        if (j + 128 < N_) __builtin_prefetch(xj + (size_t)128 * D_GAT_, 0, 1);
        float s0 = 0.f, s1 = 0.f, s2 = 0.f;
        for (int c = 0; c < 32; ++c) s0 += atts[c]      * lrelu_f(xj[c]      + xrs[c]);
        for (int c = 0; c < 32; ++c) s1 += atts[32 + c] * lrelu_f(xj[32 + c] + xrs[32 + c]);
        for (int c = 0; c < 32; ++c) s2 += atts[64 + c] * lrelu_f(xj[64 + c] + xrs[64 + c]);
        sc[j * 3 + 0] = s0; sc[j * 3 + 1] = s1; sc[j * 3 + 2] = s2;
    }
    __syncthreads();

    for (int h = 0; h < 3; ++h) {
        float m = -INFINITY;
        for (int j = tid; j < N_; j += 128) m = fmaxf(m, sc[j * 3 + h]);
        red[tid] = m; __syncthreads();
        for (int s = 64; s > 0; s >>= 1) {
            if (tid < s) red[tid] = fmaxf(red[tid], red[tid + s]);
            __syncthreads();
        }
        if (tid == 0) mh[h] = red[0];
        __syncthreads();
        float e = 0.f;
        for (int j = tid; j < N_; j += 128) e += expf(sc[j * 3 + h] - mh[h]);
        red[tid] = e; __syncthreads();
        for (int s = 64; s > 0; s >>= 1) {
            if (tid < s) red[tid] += red[tid + s];
            __syncthreads();
        }
        if (tid == 0) sh[h] = red[0];
        __syncthreads();
    }

    float outc = 0.0f;
    if (tid < D_GAT_) {
        int h = tid / 32;
        float m = mh[h], inv = 1.0f / sh[h];
        for (int j = 0; j < N_; ++j) {
            float alpha = expf(sc[j * 3 + h] - m) * inv;
            outc += alpha * X[(size_t)j * D_GAT_ + tid];
        }
        outc = gelu_f(outc + bias[tid]);
    }
    // LN over 96 channels held in threads 0..95
    red[tid] = (tid < D_GAT_) ? outc : 0.0f;
    __syncthreads();
    for (int s = 64; s > 0; s >>= 1) { if (tid < s) red[tid] += red[tid + s]; __syncthreads(); }
    float mu = red[0] / (float)D_GAT_;
    __syncthreads();
    float d = (tid < D_GAT_) ? (outc - mu) : 0.0f;
    red[tid] = d * d;
    __syncthreads();
    for (int s = 64; s > 0; s >>= 1) { if (tid < s) red[tid] += red[tid + s]; __syncthreads(); }
    float rs = rsqrtf(red[0] / (float)D_GAT_ + LN_EPS_);
    if (tid < D_GAT_)
        g0_out[b * ostride + tid] = (outc - mu) * rs * lg[tid] + lb[tid];
}

// ---------------------------------------------------------------------------
// K4: fused g1 = LN(gelu(xl1 + bias1)) (node-0 row patched with attention
// result) + conv2 lin_l: xl2 = g1 @ Wl2 + bl2. One wave per 16-node tile.
// Wlt: [96][96] col-major.
// ---------------------------------------------------------------------------
__global__ __launch_bounds__(32) void k_conv_mid(
    const float* __restrict__ xl_in, const float* __restrict__ bias_pre,
    const float* __restrict__ lg, const float* __restrict__ lb,
    const float* __restrict__ g10, const float* __restrict__ Wlt,
    const float* __restrict__ blv, float* __restrict__ out) {
    __shared__ float g[16 * D_GAT_];
    __shared__ float o[16 * D_GAT_];
    const int lane = threadIdx.x;
    const int tile = blockIdx.x;
    const int b  = tile / (N_ / 16);
    const int nt = tile % (N_ / 16);
    const int m = lane & 15, hi = lane >> 4;
    const size_t base = ((size_t)b * N_ + (size_t)nt * 16) * D_GAT_;

    for (int i = lane * 4; i < 16 * D_GAT_; i += 128) {
        v4f v = *(const v4f*)&xl_in[base + i];
        int c = i % D_GAT_;
#pragma unroll
        for (int u = 0; u < 4; ++u) v[u] = gelu_f(v[u] + bias_pre[c + u]);
        *(v4f*)&g[i] = v;
    }
    __syncthreads();
    ln_tile(g, D_GAT_, lg, lb, lane);
    __syncthreads();
    if (nt == 0) {  // node-0 row comes from the attention kernel
        for (int i = lane; i < D_GAT_; i += 32) g[i] = g10[b * D_GAT_ + i];
    }
    __syncthreads();

    for (int ntile = 0; ntile < 6; ++ntile) {
        int n0 = ntile * 16;
        const float* wc = Wlt + (size_t)(n0 + m) * D_GAT_;
        v8f acc = {};
#pragma unroll 8
        for (int k0 = 0; k0 < D_GAT_; k0 += 4) {
            int ka = k0 + 2 * hi;
            v2f a  = *(const v2f*)&g[m * D_GAT_ + ka];
            v2f bb = *(const v2f*)&wc[ka];
            acc = __builtin_amdgcn_wmma_f32_16x16x4_f32(
                false, a, false, bb, (short)0, acc, false, false);
        }
#pragma unroll
        for (int r = 0; r < 8; ++r) {
            int row = r + 8 * hi, col = n0 + m;
            o[row * D_GAT_ + col] = acc[r] + blv[col];
        }
    }
    __syncthreads();
    float* outp = out + base;
    for (int i = lane * 4; i < 16 * D_GAT_; i += 128)
        *(v4f*)&outp[i] = *(const v4f*)&o[i];
}

// ---------------------------------------------------------------------------
// K7: g2 = LN(gelu(xl2 + bias2)) for nodes > 0 (node 0 written by k_att).
// ---------------------------------------------------------------------------
__global__ __launch_bounds__(32) void k_postln(
    const float* __restrict__ xin, const float* __restrict__ bias_pre,
    const float* __restrict__ lg, const float* __restrict__ lb,
    float* __restrict__ out) {
    __shared__ float g[16 * D_GAT_];
    const int lane = threadIdx.x;
    const int tile = blockIdx.x;
    const int b  = tile / (N_ / 16);
    const int nt = tile % (N_ / 16);
    const size_t base = ((size_t)b * N_ + (size_t)nt * 16) * D_GAT_;

    for (int i = lane * 4; i < 16 * D_GAT_; i += 128) {
        v4f v = *(const v4f*)&xin[base + i];
        int c = i % D_GAT_;
#pragma unroll
        for (int u = 0; u < 4; ++u) v[u] = gelu_f(v[u] + bias_pre[c + u]);
        *(v4f*)&g[i] = v;
    }
    __syncthreads();
    ln_tile(g, D_GAT_, lg, lb, lane);
    __syncthreads();
    for (int i = lane * 4; i < 16 * D_GAT_; i += 128) {
        int r = i / D_GAT_;
        if (nt == 0 && r == 0) continue;  // keep attention-produced node-0 row
        *(v4f*)&out[base + i] = *(const v4f*)&g[i];
    }
}

// ---------------------------------------------------------------------------
// K8: per-batch per-channel softmax pooling over N, then fc_out+GELU+LN.
// ---------------------------------------------------------------------------
__global__ __launch_bounds__(128) void k_pool(
    const float* __restrict__ g2, const float* __restrict__ tptr,
    const float* __restrict__ Wout, const float* __restrict__ bout,
    const float* __restrict__ lg, const float* __restrict__ lb,
    float* __restrict__ out) {
    const int b = blockIdx.x, tid = threadIdx.x;
    __shared__ float pooled[D_GAT_];
    __shared__ float red[128];
    const float tv = tptr[0];
    const float* X = g2 + (size_t)b * N_ * D_GAT_;

    if (tid < D_GAT_) {
        float m = -INFINITY;
        for (int j = 0; j < N_; ++j) {
            if ((j & 15) == 0 && j + 16 < N_)
                __builtin_prefetch(&X[(size_t)(j + 16) * D_GAT_ + tid], 0, 1);
            m = fmaxf(m, tv * X[(size_t)j * D_GAT_ + tid]);
        }
        float s = 0.f, p = 0.f;
        for (int j = 0; j < N_; ++j) {
            float v = X[(size_t)j * D_GAT_ + tid];
            float e = expf(tv * v - m);
            s += e; p += e * v;
        }
        pooled[tid] = p / s;
    }
    __syncthreads();

    float o[2];
    for (int u = 0; u < 2; ++u) {
        int c = tid * 2 + u;
        float acc = bout[c];
        for (int k = 0; k < D_GAT_; ++k) acc += pooled[k] * Wout[k * D_OUT_ + c];
        o[u] = gelu_f(acc);
    }
    red[tid] = o[0] + o[1];
    __syncthreads();
    for (int s = 64; s > 0; s >>= 1) { if (tid < s) red[tid] += red[tid + s]; __syncthreads(); }
    float mu = red[0] / (float)D_OUT_;
    __syncthreads();
    red[tid] = (o[0] - mu) * (o[0] - mu) + (o[1] - mu) * (o[1] - mu);
    __syncthreads();
    for (int s = 64; s > 0; s >>= 1) { if (tid < s) red[tid] += red[tid + s]; __syncthreads(); }
    float rs = rsqrtf(red[0] / (float)D_OUT_ + LN_EPS_);
    for (int u = 0; u < 2; ++u) {
        int c = tid * 2 + u;
        out[(size_t)b * D_OUT_ + c] = (o[u] - mu) * rs * lg[c] + lb[c];
    }
}

extern "C" void kernel_launch(void* const* d_in, const int* in_sizes, int n_in,
                              void* d_out, int out_size, void* d_ws, size_t ws_size,
                              hipStream_t stream) {
    const float* inputs = (const float*)d_in[0];
    const float* W_node = (const float*)d_in[1];
    const float* b_node = (const float*)d_in[2];
    const float* lnn_g  = (const float*)d_in[3];
    const float* lnn_b  = (const float*)d_in[4];
    const float* Wl1    = (const float*)d_in[5];
    const float* bl1    = (const float*)d_in[6];
    const float* Wr1    = (const float*)d_in[7];
    const float* br1    = (const float*)d_in[8];
    const float* att1   = (const float*)d_in[9];
    const float* bias1  = (const float*)d_in[10];
    const float* lnc1_g = (const float*)d_in[11];
    const float* lnc1_b = (const float*)d_in[12];
    const float* Wl2    = (const float*)d_in[13];
    const float* bl2    = (const float*)d_in[14];
    const float* Wr2    = (const float*)d_in[15];
    const float* br2    = (const float*)d_in[16];
    const float* att2   = (const float*)d_in[17];
    const float* bias2  = (const float*)d_in[18];
    const float* lnc2_g = (const float*)d_in[19];
    const float* lnc2_b = (const float*)d_in[20];
    const float* t      = (const float*)d_in[21];
    const float* W_out  = (const float*)d_in[22];
    const float* b_out  = (const float*)d_in[23];
    const float* lno_g  = (const float*)d_in[24];
    const float* lno_b  = (const float*)d_in[25];
    float* outp = (float*)d_out;

    float* ws   = (float*)d_ws;
    const size_t big = (size_t)B_ * N_ * D_GAT_;  // 25,165,824 floats
    float* buf1 = ws;          // xl1, later reused for g2
    float* buf2 = buf1 + big;  // xl2
    float* h0   = buf2 + big;           // [B,128] h1 node-0 rows
    float* xr1  = h0 + B_ * D_NODE_;    // [B,96]
    float* g10  = xr1 + B_ * D_GAT_;    // [B,96] g1[b,0]
    float* xr2  = g10 + B_ * D_GAT_;    // [B,96]
    float* Wnt  = xr2 + B_ * D_GAT_;    // [128][20] col-major zero-padded
    float* Wl1t = Wnt + D_NODE_ * KPAD_;   // [96][128] col-major
    float* Wl2t = Wl1t + D_GAT_ * D_NODE_; // [96][96]  col-major

    const int TILES = B_ * (N_ / 16);  // 16384 one-wave blocks

    k_prep_w<<<(D_NODE_ * KPAD_ + 255) / 256, 256, 0, stream>>>(
        W_node, Wnt, F_IN_, KPAD_, D_NODE_);
    k_prep_w<<<(D_GAT_ * D_NODE_ + 255) / 256, 256, 0, stream>>>(
        Wl1, Wl1t, D_NODE_, D_NODE_, D_GAT_);
    k_prep_w<<<(D_GAT_ * D_GAT_ + 255) / 256, 256, 0, stream>>>(
        Wl2, Wl2t, D_GAT_, D_GAT_, D_GAT_);

    k_node_xl1<<<TILES, 32, 0, stream>>>(inputs, Wnt, b_node, lnn_g, lnn_b,
                                         Wl1t, bl1, buf1, h0);
    k_xr<<<B_, 128, 0, stream>>>(h0, D_NODE_, Wr1, br1, xr1);
    k_att<<<B_, 128, 0, stream>>>(buf1, xr1, att1, bias1, lnc1_g, lnc1_b,
                                  g10, (size_t)D_GAT_);
    k_conv_mid<<<TILES, 32, 0, stream>>>(buf1, bias1, lnc1_g, lnc1_b, g10,
                                         Wl2t, bl2, buf2);
    k_xr<<<B_, 128, 0, stream>>>(g10, D_GAT_, Wr2, br2, xr2);
    k_att<<<B_, 128, 0, stream>>>(buf2, xr2, att2, bias2, lnc2_g, lnc2_b,
                                  buf1, (size_t)N_ * D_GAT_);
    k_postln<<<TILES, 32, 0, stream>>>(buf2, bias2, lnc2_g, lnc2_b, buf1);
    k_pool<<<B_, 128, 0, stream>>>(buf1, t, W_out, b_out, lno_g, lno_b, outp);
}